// LightGCN_4140348473622
// MI455X (gfx1250) — compile-verified
//
#include <hip/hip_runtime.h>
#include <hip/hip_bf16.h>

// ---------------- problem constants (match reference) ----------------
#define N_USERS   100000
#define N_ITEMS   50000
#define N_TOTAL   (N_USERS + N_ITEMS)     // 150000
#define EMB_DIM   64
#define N_LAYERS  3

typedef __attribute__((ext_vector_type(2))) float v2f;
typedef __attribute__((ext_vector_type(8))) float v8f;

// ---------------------------------------------------------------------
// Kernel 1: X = concat(user_emb, item_emb); Y = 0.  float4 vectorized.
// nU4 = N_USERS*64/4, nTot4 = N_TOTAL*64/4.
// ---------------------------------------------------------------------
__global__ void lgcn_init_emb(const float4* __restrict__ ue,
                              const float4* __restrict__ ie,
                              float4* __restrict__ X,
                              float4* __restrict__ Y,
                              int nU4, int nTot4) {
  int i = blockIdx.x * blockDim.x + threadIdx.x;
  if (i >= nTot4) return;
  X[i] = (i < nU4) ? ue[i] : ie[i - nU4];
  Y[i] = float4{0.f, 0.f, 0.f, 0.f};
}

// ---------------------------------------------------------------------
// Kernel 2: layer-0 contribution of the accumulator:
//   accU[b] = user_emb[users[b]] ; accI[b] = item_emb[items[b]]
// i ranges over batch*16 float4s (16 float4 per 64-dim row).
// ---------------------------------------------------------------------
__global__ void lgcn_init_acc(const float4* __restrict__ ue,
                              const float4* __restrict__ ie,
                              const int* __restrict__ users,
                              const int* __restrict__ items,
                              float4* __restrict__ accU,
                              float4* __restrict__ accI,
                              int n4) {
  int i = blockIdx.x * blockDim.x + threadIdx.x;
  if (i >= n4) return;
  int b = i >> 4;
  int c = i & 15;
  accU[i] = ue[(size_t)users[b] * 16 + c];
  accI[i] = ie[(size_t)items[b] * 16 + c];
}

// ---------------------------------------------------------------------
// Kernel 3: zero a float4 buffer (next SpMM target).
// ---------------------------------------------------------------------
__global__ void lgcn_zero4(float4* __restrict__ p, int n4) {
  int i = blockIdx.x * blockDim.x + threadIdx.x;
  if (i < n4) p[i] = float4{0.f, 0.f, 0.f, 0.f};
}

// ---------------------------------------------------------------------
// Kernel 4: SpMM scatter:  Y[src[e]] += val[e] * X[dst[e]]   (64-dim rows)
//
// One wave32 per edge: lane L owns floats [2L, 2L+1] of the 256B row
// (fully coalesced b64 gather + 2 f32 atomics). Edge indices are loaded
// coalesced, 32 edges per wave, then broadcast via v_readlane.
// Grid-stride over 32-edge groups; next group's edge stream is prefetched
// with global_prefetch_b8 (gfx1250).
// ---------------------------------------------------------------------
__global__ void lgcn_spmm(const float* __restrict__ val,
                          const int*  __restrict__ src,
                          const int*  __restrict__ dst,
                          const float* __restrict__ X,
                          float* __restrict__ Y,
                          int nnz) {
  const int lane   = threadIdx.x & 31;
  const int waveId = (blockIdx.x * blockDim.x + threadIdx.x) >> 5;
  const int nWaves = (gridDim.x * blockDim.x) >> 5;
  const int nGroups = (nnz + 31) >> 5;

  for (int g = waveId; g < nGroups; g += nWaves) {
    const int base = g << 5;
    const int e = base + lane;

    // prefetch the edge stream for this wave's next group (CDNA5 prefetch)
    const int pb = base + (nWaves << 5);
    if (pb < nnz) {
      __builtin_prefetch(val + pb + lane, 0, 1);
      __builtin_prefetch(src + pb + lane, 0, 1);
      __builtin_prefetch(dst + pb + lane, 0, 1);
    }

    float v = 0.f; int s = 0, d = 0;
    if (e < nnz) { v = val[e]; s = src[e]; d = dst[e]; }
    const int cnt = (base + 32 <= nnz) ? 32 : (nnz - base);

    for (int j = 0; j < cnt; ++j) {
      const int   sj = __builtin_amdgcn_readlane(s, j);
      const int   dj = __builtin_amdgcn_readlane(d, j);
      const float vj = __builtin_bit_cast(
          float, __builtin_amdgcn_readlane(__builtin_bit_cast(int, v), j));
      const float2 xv = *(const float2*)(X + (size_t)dj * EMB_DIM + (lane << 1));
      float* o = Y + (size_t)sj * EMB_DIM + (lane << 1);
      atomicAdd(o,     vj * xv.x);   // global_atomic_add_f32 (no return)
      atomicAdd(o + 1, vj * xv.y);
    }
  }
}

// ---------------------------------------------------------------------
// Kernel 5: accumulate this layer's output at the batch rows:
//   accU[b] += Y[users[b]] ; accI[b] += Y[N_USERS + items[b]]
// ---------------------------------------------------------------------
__global__ void lgcn_accum(const float4* __restrict__ Y,
                           const int* __restrict__ users,
                           const int* __restrict__ items,
                           float4* __restrict__ accU,
                           float4* __restrict__ accI,
                           int n4) {
  int i = blockIdx.x * blockDim.x + threadIdx.x;
  if (i >= n4) return;
  int b = i >> 4;
  int c = i & 15;
  float4 yu = Y[(size_t)users[b] * 16 + c];
  float4 yi = Y[((size_t)N_USERS + items[b]) * 16 + c];
  float4 au = accU[i], ai = accI[i];
  au.x += yu.x; au.y += yu.y; au.z += yu.z; au.w += yu.w;
  ai.x += yi.x; ai.y += yi.y; ai.z += yi.z; ai.w += yi.w;
  accU[i] = au;
  accI[i] = ai;
}

// ---------------------------------------------------------------------
// Kernel 6: scores via V_WMMA_F32_16X16X4_F32 (fp32 matrix core path).
// One wave computes a 16-pair tile: D = U_tile(16x64) * I_tile^T(64x16)
// as 16 accumulating K=4 WMMAs, then scores = diag(D) / 16.
//
// A layout (16x4 f32): lane l holds A[M = l&15][K = k + 2*(l>>4) + {0,1}]
// B layout (4x16 f32): lane l holds B[K = k + 2*(l>>4) + {0,1}][N = l&15]
//   and B[k][n] = accI[b0+n][k]  ->  identical float2 addressing as A.
// D layout (16x16 f32, 8 VGPRs): lanes 0-15 hold M=g,   N=lane
//                                lanes 16-31 hold M=g+8, N=lane-16
//   => diag D[m][m]: m<8 -> lane m, c[m];  m>=8 -> lane m+16, c[m-8].
// EXEC is all-1s: exactly batch/16 full waves are launched, no divergence
// before the WMMAs.
// ---------------------------------------------------------------------
__global__ void lgcn_score_wmma(const float* __restrict__ accU,
                                const float* __restrict__ accI,
                                float* __restrict__ scores) {
  const int lane = threadIdx.x & 31;
  const int wave = (blockIdx.x * blockDim.x + threadIdx.x) >> 5;
  const int b0 = wave << 4;

  const int row  = b0 + (lane & 15);       // M for A, N for B
  const int koff = (lane >> 4) << 1;       // K sub-offset within chunk
  const float* up = accU + (size_t)row * EMB_DIM + koff;
  const float* ip = accI + (size_t)row * EMB_DIM + koff;

  v8f c = {};
#pragma unroll
  for (int k = 0; k < EMB_DIM; k += 4) {
    v2f a = *(const v2f*)(up + k);
    v2f b = *(const v2f*)(ip + k);
    // (neg_a, A, neg_b, B, c_mod, C, reuse_a, reuse_b)
    c = __builtin_amdgcn_wmma_f32_16x16x4_f32(false, a, false, b,
                                              (short)0, c, false, false);
  }

  // extract diagonal of D
  float dval = 0.f;
#pragma unroll
  for (int g = 0; g < 8; ++g) {
    float el = c[g];
    if (lane == g || lane == g + 24) dval = el;
  }
  // light_out = acc/4 on both sides -> scores = dot/16
  if (lane < 8)        scores[b0 + lane]        = dval * 0.0625f;
  else if (lane >= 24) scores[b0 + lane - 16]   = dval * 0.0625f;
}

// ---------------------------------------------------------------------
// Host-side launcher
// ---------------------------------------------------------------------
extern "C" void kernel_launch(void* const* d_in, const int* in_sizes, int n_in,
                              void* d_out, int out_size, void* d_ws, size_t ws_size,
                              hipStream_t stream) {
  const float* user_emb  = (const float*)d_in[0];
  const float* item_emb  = (const float*)d_in[1];
  const float* graph_val = (const float*)d_in[2];
  const int*   graph_src = (const int*)d_in[3];
  const int*   graph_dst = (const int*)d_in[4];
  const int*   users     = (const int*)d_in[5];
  const int*   items     = (const int*)d_in[6];
  float*       scores    = (float*)d_out;

  const int nnz   = in_sizes[2];
  const int batch = in_sizes[5];

  // workspace: X (ping) | Y (pong) | accU | accI
  const size_t embBytes = (size_t)N_TOTAL * EMB_DIM * sizeof(float);   // 38.4 MB
  const size_t accBytes = (size_t)batch * EMB_DIM * sizeof(float);     // 1 MB
  char* ws = (char*)d_ws;
  float* X    = (float*)(ws);
  float* Y    = (float*)(ws + embBytes);
  float* accU = (float*)(ws + 2 * embBytes);
  float* accI = (float*)(ws + 2 * embBytes + accBytes);

  const int nU4   = N_USERS * EMB_DIM / 4;   // 1.6M float4
  const int nTot4 = N_TOTAL * EMB_DIM / 4;   // 2.4M float4
  const int acc4  = batch * EMB_DIM / 4;     // 65536 float4

  const int TB = 256;

  // init: X = concat(E), Y = 0, acc = layer-0 gathered rows
  lgcn_init_emb<<<(nTot4 + TB - 1) / TB, TB, 0, stream>>>(
      (const float4*)user_emb, (const float4*)item_emb,
      (float4*)X, (float4*)Y, nU4, nTot4);
  lgcn_init_acc<<<(acc4 + TB - 1) / TB, TB, 0, stream>>>(
      (const float4*)user_emb, (const float4*)item_emb,
      users, items, (float4*)accU, (float4*)accI, acc4);

  float* cur = X;
  float* nxt = Y;
  const int spmmBlocks = 2048;   // 16384 wave32s grid-striding 200K edge-groups
  for (int layer = 0; layer < N_LAYERS; ++layer) {
    if (layer > 0) {
      lgcn_zero4<<<(nTot4 + TB - 1) / TB, TB, 0, stream>>>((float4*)nxt, nTot4);
    }
    lgcn_spmm<<<spmmBlocks, TB, 0, stream>>>(graph_val, graph_src, graph_dst,
                                             cur, nxt, nnz);
    lgcn_accum<<<(acc4 + TB - 1) / TB, TB, 0, stream>>>(
        (const float4*)nxt, users, items, (float4*)accU, (float4*)accI, acc4);
    float* t = cur; cur = nxt; nxt = t;
  }

  // scores: batch/16 tiles, 1 wave each, 8 waves per block
  const int tiles = batch / 16;                 // 256
  const int scoreThreads = tiles * 32;          // 8192
  lgcn_score_wmma<<<scoreThreads / TB, TB, 0, stream>>>(accU, accI, scores);
}